// Attention_46437186404686
// MI455X (gfx1250) — compile-verified
//
#include <hip/hip_runtime.h>
#include <hip/hip_bf16.h>

typedef __attribute__((ext_vector_type(2))) float v2f;
typedef __attribute__((ext_vector_type(8))) float v8f;

// D(16x16,f32) = A(16x4,f32) * B(4x16,f32) + C   -- V_WMMA_F32_16X16X4_F32
static __device__ __forceinline__ v8f wmma4(v2f a, v2f b, v8f c) {
    return __builtin_amdgcn_wmma_f32_16x16x4_f32(
        /*neg_a=*/false, a, /*neg_b=*/false, b,
        /*c_mod=*/(short)0, c, /*reuse_a=*/false, /*reuse_b=*/false);
}

// ---------------------------------------------------------------------------
// Kernel 1: fused QKV projection.  C[m][n] = sum_k x[m][k] * W[n][k]
// M = 32 (batch), N = 4096(q)+1024(k)+1024(v), K = 4096.
// One wave per 16-wide N tile; B-frag (weights) loaded once, used by both
// M tiles -> weight bytes stream exactly once from HBM.
// ---------------------------------------------------------------------------
__global__ __launch_bounds__(32) void qkv_proj_kernel(
    const float* __restrict__ x,
    const float* __restrict__ wq, const float* __restrict__ wk,
    const float* __restrict__ wv,
    float* __restrict__ qo, float* __restrict__ ko, float* __restrict__ vo)
{
    const int nt   = blockIdx.x;          // 0..383
    const int lane = threadIdx.x;         // 0..31
    const int half = lane >> 4;           // K-half select per ISA layout
    const int col  = lane & 15;           // N (and A-row) index

    const float* w; float* out; int ostride; int nbase;
    if (nt < 256)      { w = wq; out = qo; ostride = 4096; nbase = nt * 16; }
    else if (nt < 320) { w = wk; out = ko; ostride = 1024; nbase = (nt - 256) * 16; }
    else               { w = wv; out = vo; ostride = 1024; nbase = (nt - 320) * 16; }

    const float* wrow = w + (size_t)(nbase + col) * 4096;  // B col -> W row (K-contig)
    const float* xr0  = x + (size_t)col * 4096;            // A rows, M-tile 0
    const float* xr1  = x + (size_t)(col + 16) * 4096;     // A rows, M-tile 1

    v8f c0 = {}; v8f c1 = {};
    for (int k0 = 0; k0 < 4096; k0 += 64) {
        __builtin_prefetch(wrow + k0 + 512, 0, 1);         // global_prefetch_b8
#pragma unroll
        for (int k4 = 0; k4 < 64; k4 += 4) {
            const int kk = k0 + k4 + 2 * half;
            v2f a0 = *(const v2f*)(xr0 + kk);
            v2f a1 = *(const v2f*)(xr1 + kk);
            v2f bb = *(const v2f*)(wrow + kk);
            c0 = wmma4(a0, bb, c0);
            c1 = wmma4(a1, bb, c1);
        }
    }
#pragma unroll
    for (int v = 0; v < 8; ++v) {
        const int m = v + 8 * half;
        out[(size_t)m        * ostride + nbase + col] = c0[v];
        out[(size_t)(m + 16) * ostride + nbase + col] = c1[v];
    }
}

// ---------------------------------------------------------------------------
// Kernel 2: RoPE on q (with 1/sqrt(128) folded in) and k, in-place in ws.
// ---------------------------------------------------------------------------
__global__ __launch_bounds__(256) void rope_kernel(
    const float* __restrict__ cvec, const float* __restrict__ svec,
    float* __restrict__ q, float* __restrict__ k)
{
    const int i = blockIdx.x * 256 + threadIdx.x;
    const float scale = 0.08838834764831845f;  // 1/sqrt(128)
    if (i < 32 * 32 * 64) {                    // q pairs
        const int p  = i & 63;
        const int hb = i >> 6;                 // b*32 + head
        float* t = q + (size_t)hb * 128 + 2 * p;
        const float c = cvec[p], s = svec[p];
        const float a = t[0], b = t[1];
        t[0] = (a * c - b * s) * scale;
        t[1] = (a * s + b * c) * scale;
    } else {                                   // k pairs
        const int i2 = i - 32 * 32 * 64;
        const int p  = i2 & 63;
        const int hb = i2 >> 6;                // b*8 + kv-head
        float* t = k + (size_t)hb * 128 + 2 * p;
        const float c = cvec[p], s = svec[p];
        const float a = t[0], b = t[1];
        t[0] = a * c - b * s;
        t[1] = a * s + b * c;
    }
}

// ---------------------------------------------------------------------------
// Kernel 3: attention for one (batch, kv-head) per 256-thread block (8 waves).
// Phase 1: S[4][4096] = q.K^T via WMMA (rows 4..15 padded, discarded).
// Phase 2: block softmax.
// Phase 3: O = P.V via WMMA, with V staged through LDS using double-buffered
//          GLOBAL_LOAD_ASYNC_TO_LDS_B128 (ASYNCcnt) -- coalesced 16B/lane
//          copies instead of strided scalar gathers. V rows padded to 136
//          floats so the two rows (delta-t = 2) touched by one ds_load_b32
//          land on disjoint LDS banks (2*136 mod 64 = 16).
// Position 4095 comes from the freshly projected k/v in ws, not the cache.
// ---------------------------------------------------------------------------
#define VCHUNK 32            // t-rows staged per chunk
#define VPITCH 136           // padded floats per staged V row

__global__ __launch_bounds__(256) void attn_kernel(
    const float* __restrict__ cache_k, const float* __restrict__ cache_v,
    const float* __restrict__ qws, const float* __restrict__ kws,
    const float* __restrict__ vws, float* __restrict__ aout)
{
    __shared__ float S[4 * 4096];              // scores / probs      (64 KB)
    __shared__ float qs[4 * 128];              // q rows              ( 2 KB)
    __shared__ float red[256];                 // reduction scratch   ( 1 KB)
    __shared__ float rinv[4];                  // 1 / row-sum
    __shared__ float Vs[2][VCHUNK * VPITCH];   // V staging           (34 KB)

    const int bg = blockIdx.x;
    const int b  = bg >> 3;
    const int g  = bg & 7;
    const int tid  = threadIdx.x;
    const int wave = tid >> 5;
    const int lane = tid & 31;
    const int half = lane >> 4;
    const int col  = lane & 15;

    for (int i = tid; i < 512; i += 256)
        qs[i] = qws[(size_t)b * 4096 + g * 512 + i];
    __syncthreads();

    // Hoist A-frags (q) for the whole score K-loop: 32 x float2 per lane.
    v2f areg[32];
    {
        const int r = col & 3;
#pragma unroll
        for (int k4 = 0; k4 < 32; ++k4)
            areg[k4] = *(const v2f*)(&qs[r * 128 + k4 * 4 + 2 * half]);
    }

    const float* kbase = cache_k + (((size_t)b * 4096) * 8 + g) * 128; // + t*1024
    const float* knew  = kws + (size_t)b * 1024 + g * 128;

    // ---- Phase 1: scores. Each wave does 32 t-tiles of 16 positions. ----
    for (int tt = wave * 32; tt < wave * 32 + 32; ++tt) {
        const int t0 = tt * 16;
        const int tr = t0 + col;                    // key row this lane feeds
        const float* krow = (tr < 4095) ? (kbase + (size_t)tr * 1024) : knew;
        v8f c = {};
#pragma unroll
        for (int k4 = 0; k4 < 32; ++k4) {
            v2f bb = *(const v2f*)(krow + k4 * 4 + 2 * half);
            c = wmma4(areg[k4], bb, c);
        }
        if (half == 0) {                            // rows m = v (v<4) valid
#pragma unroll
            for (int v = 0; v < 4; ++v) S[v * 4096 + t0 + col] = c[v];
        }
    }
    __syncthreads();

    // ---- Phase 2: softmax over 4096 per row; 64 threads per row. ----
    const int r = tid >> 6;
    const int j = tid & 63;
    float m = -3.0e38f;
    for (int i = 0; i < 64; ++i) m = fmaxf(m, S[r * 4096 + j + i * 64]);
    red[tid] = m;
    __syncthreads();
    for (int off = 32; off; off >>= 1) {
        if (j < off) red[tid] = fmaxf(red[tid], red[tid + off]);
        __syncthreads();
    }
    const float rowmax = red[r << 6];
    __syncthreads();
    float s = 0.0f;
    for (int i = 0; i < 64; ++i) {
        const int idx = r * 4096 + j + i * 64;
        const float e = __expf(S[idx] - rowmax);
        S[idx] = e;
        s += e;
    }
    red[tid] = s;
    __syncthreads();
    for (int off = 32; off; off >>= 1) {
        if (j < off) red[tid] += red[tid + off];
        __syncthreads();
    }
    if (j == 0) rinv[r] = 1.0f / red[r << 6];
    __syncthreads();

    // ---- Phase 3: O = P * V with async-staged V tiles. ----
    const float* vbase = cache_v + (((size_t)b * 4096) * 8 + g) * 128;
    const float* vnew  = vws + (size_t)b * 1024 + g * 128;
    const int d0 = wave * 16;
    const int pr = col & 3;
    v8f o = {};

    // Each wave stages VCHUNK/8 = 4 rows per chunk, 16 B per lane.
    auto stage = [&](int c, int buf) {
#pragma unroll
        for (int i = 0; i < VCHUNK / 8; ++i) {
            const int rr = wave * (VCHUNK / 8) + i;       // row within chunk
            const int tr = c * VCHUNK + rr;               // global t
            const float* src = (tr < 4095) ? (vbase + (size_t)tr * 1024) : vnew;
            const unsigned lds =
                (unsigned)(uintptr_t)(&Vs[buf][rr * VPITCH]) + (unsigned)lane * 16u;
            const unsigned long long ga =
                (unsigned long long)(uintptr_t)(src + lane * 4);
            asm volatile("global_load_async_to_lds_b128 %0, %1, off"
                         :: "v"(lds), "v"(ga) : "memory");
        }
    };

    stage(0, 0);
    for (int c = 0; c < 4096 / VCHUNK; ++c) {
        const int buf = c & 1;
        if (c + 1 < 4096 / VCHUNK) {
            stage(c + 1, buf ^ 1);
            asm volatile("s_wait_asynccnt 0x4" ::: "memory");  // chunk c landed
        } else {
            asm volatile("s_wait_asynccnt 0x0" ::: "memory");
        }
        __syncthreads();                                       // visible to all waves

        const float* vt = &Vs[buf][0];
#pragma unroll
        for (int t4 = 0; t4 < VCHUNK / 4; ++t4) {
            const int r0 = t4 * 4 + 2 * half;                  // local k rows
            v2f af = *(const v2f*)(&S[pr * 4096 + c * VCHUNK + t4 * 4 + 2 * half]);
            v2f bf;
            bf[0] = vt[(r0    ) * VPITCH + d0 + col];
            bf[1] = vt[(r0 + 1) * VPITCH + d0 + col];
            o = wmma4(af, bf, o);
        }
        __syncthreads();                 // all consumed before buffer is re-staged
    }

    if (half == 0) {
#pragma unroll
        for (int v = 0; v < 4; ++v)
            aout[(size_t)b * 4096 + g * 512 + v * 128 + d0 + col] = o[v] * rinv[v];
    }
}

// ---------------------------------------------------------------------------
// Kernel 4: output projection. out[b][d] = sum_e attn[b][e] * wo[d][e]
// M = 32, N = 4096, K = 4096. Same scheme as kernel 1.
// ---------------------------------------------------------------------------
__global__ __launch_bounds__(32) void oproj_kernel(
    const float* __restrict__ attn, const float* __restrict__ wo,
    float* __restrict__ out)
{
    const int nt   = blockIdx.x;          // 0..255
    const int lane = threadIdx.x;
    const int half = lane >> 4;
    const int col  = lane & 15;

    const float* wrow = wo + (size_t)(nt * 16 + col) * 4096;
    const float* a0   = attn + (size_t)col * 4096;
    const float* a1   = attn + (size_t)(col + 16) * 4096;

    v8f c0 = {}; v8f c1 = {};
    for (int k0 = 0; k0 < 4096; k0 += 64) {
        __builtin_prefetch(wrow + k0 + 512, 0, 1);
#pragma unroll
        for (int k4 = 0; k4 < 64; k4 += 4) {
            const int kk = k0 + k4 + 2 * half;
            v2f x0 = *(const v2f*)(a0 + kk);
            v2f x1 = *(const v2f*)(a1 + kk);
            v2f bb = *(const v2f*)(wrow + kk);
            c0 = wmma4(x0, bb, c0);
            c1 = wmma4(x1, bb, c1);
        }
    }
#pragma unroll
    for (int v = 0; v < 8; ++v) {
        const int mrow = v + 8 * half;
        out[(size_t)mrow        * 4096 + nt * 16 + col] = c0[v];
        out[(size_t)(mrow + 16) * 4096 + nt * 16 + col] = c1[v];
    }
}

// ---------------------------------------------------------------------------
extern "C" void kernel_launch(void* const* d_in, const int* in_sizes, int n_in,
                              void* d_out, int out_size, void* d_ws, size_t ws_size,
                              hipStream_t stream) {
    (void)in_sizes; (void)n_in; (void)out_size; (void)ws_size;
    const float* x  = (const float*)d_in[0];
    // d_in[1] = start_pos (int scalar) -- fixed at 4095 by the problem shape
    const float* fc = (const float*)d_in[2];
    const float* fs = (const float*)d_in[3];
    const float* ck = (const float*)d_in[4];
    const float* cv = (const float*)d_in[5];
    const float* wq = (const float*)d_in[6];
    const float* wk = (const float*)d_in[7];
    const float* wv = (const float*)d_in[8];
    const float* wo = (const float*)d_in[9];
    float* out = (float*)d_out;

    float* ws   = (float*)d_ws;
    float* q    = ws;             // 32*32*128 = 131072 floats
    float* k    = ws + 131072;    // 32*8*128  =  32768 floats
    float* v    = ws + 163840;    // 32*8*128  =  32768 floats
    float* attn = ws + 196608;    // 32*32*128 = 131072 floats

    qkv_proj_kernel<<<384, 32, 0, stream>>>(x, wq, wk, wv, q, k, v);
    rope_kernel<<<320, 256, 0, stream>>>(fc, fs, q, k);
    attn_kernel<<<256, 256, 0, stream>>>(ck, cv, q, k, v, attn);
    oproj_kernel<<<256, 32, 0, stream>>>(attn, wo, out);
}